// GenAttnDecoderRNN_69114613728295
// MI455X (gfx1250) — compile-verified
//
#include <hip/hip_runtime.h>
#include <math.h>

#define HDIM 1024
#define H2   2048
#define SDIM 2048
#define VDIM 50257
#define VPAD 50304   // 393 blocks * 128 rows, multiple of 16

typedef float v2f __attribute__((ext_vector_type(2)));
typedef float v8f __attribute__((ext_vector_type(8)));

__device__ __forceinline__ float waveReduceSum(float v) {
#pragma unroll
  for (int off = 16; off > 0; off >>= 1) v += __shfl_xor(v, off, 32);
  return v;
}

// ---------------------------------------------------------------- kernel 1
// x = [emb[input_id], last_context]  -> ws_x [2H]
__global__ void k_prepare(const int* __restrict__ ids,
                          const float* __restrict__ emb,
                          const float* __restrict__ lastCtx,
                          float* __restrict__ x) {
  int t = blockIdx.x * 256 + threadIdx.x;
  int id = ids[0];
  if (t < HDIM)      x[t] = emb[(size_t)id * HDIM + t];
  else if (t < H2)   x[t] = lastCtx[t - HDIM];
}

// ---------------------------------------------------------------- kernel 2
// One wave per output element j: six row-dot-products + GRU gates.
__global__ void k_gru(const float* __restrict__ x,      // [2H]
                      const float* __restrict__ hprev,  // [H]
                      const float* __restrict__ Wih,    // [3H,2H]
                      const float* __restrict__ Whh,    // [3H,H]
                      const float* __restrict__ bih,
                      const float* __restrict__ bhh,
                      float* __restrict__ rOut,         // ws_r [H]
                      float* __restrict__ hidOut) {     // d_out hidden [H]
  int lane = threadIdx.x & 31;
  int wave = threadIdx.x >> 5;
  int j = blockIdx.x * 8 + wave;
  if (j >= HDIM) return;

  const float4* x4  = (const float4*)x;
  const float4* h4  = (const float4*)hprev;
  const float4* wir = (const float4*)(Wih + (size_t)j * H2);
  const float4* wiz = (const float4*)(Wih + (size_t)(j + HDIM) * H2);
  const float4* win = (const float4*)(Wih + (size_t)(j + 2 * HDIM) * H2);
  const float4* whr = (const float4*)(Whh + (size_t)j * HDIM);
  const float4* whz = (const float4*)(Whh + (size_t)(j + HDIM) * HDIM);
  const float4* whn = (const float4*)(Whh + (size_t)(j + 2 * HDIM) * HDIM);

  float air = 0.f, aiz = 0.f, ain = 0.f;
  for (int i = lane; i < H2 / 4; i += 32) {
    float4 xv = x4[i];
    float4 a = wir[i], b = wiz[i], c = win[i];
    air += a.x * xv.x + a.y * xv.y + a.z * xv.z + a.w * xv.w;
    aiz += b.x * xv.x + b.y * xv.y + b.z * xv.z + b.w * xv.w;
    ain += c.x * xv.x + c.y * xv.y + c.z * xv.z + c.w * xv.w;
  }
  float ahr = 0.f, ahz = 0.f, ahn = 0.f;
  for (int i = lane; i < HDIM / 4; i += 32) {
    float4 hv = h4[i];
    float4 a = whr[i], b = whz[i], c = whn[i];
    ahr += a.x * hv.x + a.y * hv.y + a.z * hv.z + a.w * hv.w;
    ahz += b.x * hv.x + b.y * hv.y + b.z * hv.z + b.w * hv.w;
    ahn += c.x * hv.x + c.y * hv.y + c.z * hv.z + c.w * hv.w;
  }
  air = waveReduceSum(air); aiz = waveReduceSum(aiz); ain = waveReduceSum(ain);
  ahr = waveReduceSum(ahr); ahz = waveReduceSum(ahz); ahn = waveReduceSum(ahn);

  if (lane == 0) {
    float gr  = air + bih[j]          + ahr + bhh[j];
    float gz  = aiz + bih[j + HDIM]   + ahz + bhh[j + HDIM];
    float gni = ain + bih[j + 2*HDIM];
    float gnh = ahn + bhh[j + 2*HDIM];
    float rg = 1.f / (1.f + expf(-gr));
    float zg = 1.f / (1.f + expf(-gz));
    float ng = tanhf(gni + rg * gnh);
    float hv = hprev[j];
    float o  = (1.f - zg) * ng + zg * hv;
    rOut[j]   = o;
    hidOut[j] = o;
  }
}

// ---------------------------------------------------------------- kernel 3
// Partial u: slice `by` covers 128 j's. up[by*H + k] = sum_j W[j,k]*r[j].
// Block bx==4, by==0 computes c0 = b.r. Grid dim3(5, 8).
__global__ void k_attn_u(const float* __restrict__ attn_W,
                         const float* __restrict__ attn_b,
                         const float* __restrict__ r,
                         float* __restrict__ up,
                         float* __restrict__ c0) {
  if (blockIdx.x < 4) {
    int k  = blockIdx.x * 256 + threadIdx.x;
    int j0 = blockIdx.y * 128;
    float acc = 0.f;
    for (int j = j0; j < j0 + 128; ++j)
      acc += attn_W[(size_t)j * HDIM + k] * r[j];   // coalesced over k
    up[blockIdx.y * HDIM + k] = acc;
  } else if (blockIdx.y == 0) {
    __shared__ float red[256];
    float acc = 0.f;
    for (int j = threadIdx.x; j < HDIM; j += 256) acc += attn_b[j] * r[j];
    red[threadIdx.x] = acc; __syncthreads();
    for (int s = 128; s > 0; s >>= 1) {
      if (threadIdx.x < s) red[threadIdx.x] += red[threadIdx.x + s];
      __syncthreads();
    }
    if (threadIdx.x == 0) *c0 = red[0];
  }
}

// u[k] = sum of 8 partial slices (deterministic order)
__global__ void k_u_reduce(const float* __restrict__ up, float* __restrict__ u) {
  int k = blockIdx.x * 256 + threadIdx.x;
  float acc = 0.f;
#pragma unroll
  for (int s = 0; s < 8; ++s) acc += up[s * HDIM + k];
  u[k] = acc;
}

// ---------------------------------------------------------------- kernel 4
// energies[s] = E[s].u + c0   (one wave per s)
__global__ void k_energy(const float* __restrict__ E,
                         const float* __restrict__ u,
                         const float* __restrict__ c0,
                         float* __restrict__ energ) {
  int lane = threadIdx.x & 31;
  int wave = threadIdx.x >> 5;
  int s = blockIdx.x * 8 + wave;            // 256 blocks * 8 = 2048 = S
  const float4* e4 = (const float4*)(E + (size_t)s * HDIM);
  const float4* u4 = (const float4*)u;
  float acc = 0.f;
  for (int i = lane; i < HDIM / 4; i += 32) {
    float4 a = e4[i], b = u4[i];
    acc += a.x * b.x + a.y * b.y + a.z * b.z + a.w * b.w;
  }
  acc = waveReduceSum(acc);
  if (lane == 0) energ[s] = acc + *c0;
}

// ---------------------------------------------------------------- kernel 5
// softmax over S=2048, one block of 1024 (2 elems/thread)
__global__ void k_attn_softmax(const float* __restrict__ energ,
                               float* __restrict__ wOut) {
  __shared__ float red[1024];
  int t = threadIdx.x;
  float a = energ[t], b = energ[t + 1024];
  red[t] = fmaxf(a, b); __syncthreads();
  for (int s = 512; s > 0; s >>= 1) {
    if (t < s) red[t] = fmaxf(red[t], red[t + s]);
    __syncthreads();
  }
  float M = red[0]; __syncthreads();
  float ea = expf(a - M), eb = expf(b - M);
  red[t] = ea + eb; __syncthreads();
  for (int s = 512; s > 0; s >>= 1) {
    if (t < s) red[t] += red[t + s];
    __syncthreads();
  }
  float inv = 1.f / red[0];
  wOut[t]        = ea * inv;
  wOut[t + 1024] = eb * inv;
}

// ---------------------------------------------------------------- kernel 6
// Partial context: slice `by` covers 256 s. cp[by*H + k] = sum_s w[s]*E[s,k].
__global__ void k_context(const float* __restrict__ E,
                          const float* __restrict__ w,
                          float* __restrict__ cp) {
  int k  = blockIdx.x * 256 + threadIdx.x;
  int s0 = blockIdx.y * 256;
  float acc = 0.f;
  for (int s = s0; s < s0 + 256; ++s)
    acc += w[s] * E[(size_t)s * HDIM + k];
  cp[blockIdx.y * HDIM + k] = acc;
}

__global__ void k_ctx_reduce(const float* __restrict__ cp, float* __restrict__ ctx) {
  int k = blockIdx.x * 256 + threadIdx.x;
  float acc = 0.f;
#pragma unroll
  for (int s = 0; s < 8; ++s) acc += cp[s * HDIM + k];
  ctx[k] = acc;
}

// ---------------------------------------------------------------- kernel 7
// logits[v] = out_W[v,:] . [r,ctx]  via V_WMMA_F32_16X16X4_F32.
// One wave owns a 16-row tile; A = 16x4 f32 chunk of out_W (each lane one
// b64 load => exactly the 256B/step the wave consumes), B = y-chunk broadcast
// across all 16 N columns (same K map as A: K = 2*half + vgpr), C accumulates.
// Stores are branch-free b128 into VPAD-padded workspace; bias added later.
__global__ void k_logits(const float* __restrict__ r,
                         const float* __restrict__ ctx,
                         const float* __restrict__ out_W,
                         float* __restrict__ logits) {   // [VPAD]
  __shared__ float y[H2];
  for (int i = threadIdx.x; i < H2; i += 256)
    y[i] = (i < HDIM) ? r[i] : ctx[i - HDIM];
  __syncthreads();

  int lane = threadIdx.x & 31;
  int wave = threadIdx.x >> 5;
  int tile = blockIdx.x * 8 + wave;
  int rowBase = tile * 16;                // < VPAD always (393*8 tiles exact)

  int hi = lane >> 4;                     // lane half selects K offset {0,2}
  int m  = lane & 15;                     // A row within tile
  int rowA = rowBase + m;
  if (rowA > VDIM - 1) rowA = VDIM - 1;   // clamp ragged tile (loads only)
  const float* wrow = out_W + (size_t)rowA * H2 + 2 * hi;

#if __has_builtin(__builtin_amdgcn_wmma_f32_16x16x4_f32)
  v8f c = {};
#pragma unroll 4
  for (int k0 = 0; k0 < H2; k0 += 4) {
    v2f a = *(const v2f*)(wrow + k0);           // global_load_b64
    v2f b = *(const v2f*)(&y[k0 + 2 * hi]);     // ds_load_b64 (broadcast)
    c = __builtin_amdgcn_wmma_f32_16x16x4_f32(false, a, false, b,
                                              (short)0, c, false, false);
  }
  // All N columns of D are identical; lanes with N==0 hold the answer.
  // C/D layout: lanes 0-15 -> rows M=vgpr (0..7), lanes 16-31 -> M=vgpr+8.
  if (m == 0) {
    int base = rowBase + hi * 8;                // 32B-aligned, always in VPAD
    float4 lo, hiq;
    lo.x  = c[0]; lo.y  = c[1]; lo.z  = c[2]; lo.w  = c[3];
    hiq.x = c[4]; hiq.y = c[5]; hiq.z = c[6]; hiq.w = c[7];
    *(float4*)(logits + base)     = lo;         // global_store_b128
    *(float4*)(logits + base + 4) = hiq;
  }
#else
  // Fallback: wave-cooperative dot per row (still b128 streaming).
  for (int r16 = 0; r16 < 16; ++r16) {
    int row = rowBase + r16;
    const float4* w4 = (const float4*)(out_W + (size_t)((row < VDIM) ? row : VDIM - 1) * H2);
    const float4* y4 = (const float4*)y;
    float acc = 0.f;
    for (int i = lane; i < H2 / 4; i += 32) {
      float4 a = w4[i], b = y4[i];
      acc += a.x * b.x + a.y * b.y + a.z * b.z + a.w * b.w;
    }
    acc = waveReduceSum(acc);
    if (lane == 0) logits[row] = acc;
  }
#endif
}

// ---------------------------------------------------------------- kernel 8
// Online log-sum-exp over x[i] = logits[i] + out_b[i], 64 blocks.
__device__ __forceinline__ void lse_merge(float& m, float& s, float m2, float s2) {
  float M = fmaxf(m, m2);
  s = s * expf(m - M) + s2 * expf(m2 - M);
  m = M;
}

__global__ void k_lse_partial(const float* __restrict__ lg,
                              const float* __restrict__ b,
                              float* __restrict__ pairs) {  // [64][2]
  __shared__ float rm[256], rs[256];
  int t = threadIdx.x;
  float m = -3.4e38f, s = 0.f;
  for (int i = blockIdx.x * 256 + t; i < VDIM; i += 64 * 256) {
    float x = lg[i] + b[i];
    float M = fmaxf(m, x);
    s = s * expf(m - M) + expf(x - M);
    m = M;
  }
  rm[t] = m; rs[t] = s; __syncthreads();
  for (int k = 128; k > 0; k >>= 1) {
    if (t < k) {
      float mm = rm[t], ss = rs[t];
      lse_merge(mm, ss, rm[t + k], rs[t + k]);
      rm[t] = mm; rs[t] = ss;
    }
    __syncthreads();
  }
  if (t == 0) { pairs[blockIdx.x * 2] = rm[0]; pairs[blockIdx.x * 2 + 1] = rs[0]; }
}

__global__ void k_lse_final(const float* __restrict__ pairs,
                            float* __restrict__ lz) {       // 1 block, 64 thr
  __shared__ float rm[64], rs[64];
  int t = threadIdx.x;
  rm[t] = pairs[t * 2]; rs[t] = pairs[t * 2 + 1]; __syncthreads();
  for (int k = 32; k > 0; k >>= 1) {
    if (t < k) {
      float mm = rm[t], ss = rs[t];
      lse_merge(mm, ss, rm[t + k], rs[t + k]);
      rm[t] = mm; rs[t] = ss;
    }
    __syncthreads();
  }
  if (t == 0) *lz = rm[0] + logf(rs[0]);
}

__global__ void k_lsm_write(const float* __restrict__ lg,
                            const float* __restrict__ b,
                            const float* __restrict__ lz,
                            float* __restrict__ out) {
  int i = blockIdx.x * 256 + threadIdx.x;
  if (i < VDIM) out[i] = lg[i] + b[i] - *lz;
}

// ----------------------------------------------------------------
extern "C" void kernel_launch(void* const* d_in, const int* in_sizes, int n_in,
                              void* d_out, int out_size, void* d_ws, size_t ws_size,
                              hipStream_t stream) {
  (void)in_sizes; (void)n_in; (void)out_size; (void)ws_size;
  const int*   input_ids    = (const int*)  d_in[0];
  const float* last_context = (const float*)d_in[1];
  const float* last_hidden  = (const float*)d_in[2];
  const float* enc          = (const float*)d_in[3];
  const float* emb          = (const float*)d_in[4];
  const float* attn_W       = (const float*)d_in[5];
  const float* attn_b       = (const float*)d_in[6];
  const float* gru_Wih      = (const float*)d_in[7];
  const float* gru_Whh      = (const float*)d_in[8];
  const float* gru_bih      = (const float*)d_in[9];
  const float* gru_bhh      = (const float*)d_in[10];
  const float* out_W        = (const float*)d_in[11];
  const float* out_b        = (const float*)d_in[12];

  float* out   = (float*)d_out;
  float* o_log = out;                      // [V]
  float* o_ctx = out + VDIM;               // [H]
  float* o_hid = out + VDIM + HDIM;        // [H]
  float* o_att = out + VDIM + 2 * HDIM;    // [S]

  float* ws      = (float*)d_ws;
  float* ws_x    = ws;               // 2048
  float* ws_r    = ws + 2048;        // 1024
  float* ws_u    = ws + 3072;        // 1024
  float* ws_c0   = ws + 4096;        // 1 (padded to 8)
  float* ws_e    = ws + 4104;        // 2048
  float* ws_part = ws + 6152;        // 8192 (shared: u-partials then ctx-partials)
  float* ws_pair = ws + 14344;       // 128
  float* ws_lz   = ws + 14472;       // 1 (padded to 8)
  float* ws_lg   = ws + 14480;       // VPAD (50304)

  k_prepare<<<8, 256, 0, stream>>>(input_ids, emb, last_context, ws_x);
  k_gru<<<128, 256, 0, stream>>>(ws_x, last_hidden, gru_Wih, gru_Whh,
                                 gru_bih, gru_bhh, ws_r, o_hid);
  k_attn_u<<<dim3(5, 8), 256, 0, stream>>>(attn_W, attn_b, ws_r, ws_part, ws_c0);
  k_u_reduce<<<4, 256, 0, stream>>>(ws_part, ws_u);
  k_energy<<<256, 256, 0, stream>>>(enc, ws_u, ws_c0, ws_e);
  k_attn_softmax<<<1, 1024, 0, stream>>>(ws_e, o_att);
  k_context<<<dim3(4, 8), 256, 0, stream>>>(enc, o_att, ws_part);
  k_ctx_reduce<<<4, 256, 0, stream>>>(ws_part, o_ctx);
  k_logits<<<VPAD / 128, 256, 0, stream>>>(ws_r, o_ctx, out_W, ws_lg);
  k_lse_partial<<<64, 256, 0, stream>>>(ws_lg, out_b, ws_pair);
  k_lse_final<<<1, 64, 0, stream>>>(ws_pair, ws_lz);
  k_lsm_write<<<(VDIM + 255) / 256, 256, 0, stream>>>(ws_lg, out_b, ws_lz, o_log);
}